// Model_45251775430770
// MI455X (gfx1250) — compile-verified
//
#include <hip/hip_runtime.h>
#include <math.h>

typedef __attribute__((ext_vector_type(2))) float v2f;
typedef __attribute__((ext_vector_type(8))) float v8f;

#define N_NODES 256
#define WSTR 66      // padded stride for 64-wide LDS matrices (bank-conflict avoidance)
#define RSTR 256     // x-row stride in LDS

__device__ __forceinline__ void wave_wait_lds() {
    // order wave-private cross-lane LDS store->load; also a compiler memory barrier
    asm volatile("s_wait_dscnt 0" ::: "memory");
}

// tanh via native v_exp_f32: tanh(x) = 1 - 2/(exp(2x)+1)
// exp->inf  => 1-0 = 1 ; exp->0 => 1-2 = -1 ; no NaN at saturation
__device__ __forceinline__ float fast_tanh(float x) {
    return 1.f - 2.f / (__expf(2.f * x) + 1.f);
}
__device__ __forceinline__ float fast_sigmoid(float x) {
    return 1.f / (1.f + __expf(-x));
}

// ---------------------------------------------------------------------------
// Kernel 1: deterministic subgraph selection (1 wave).
// top-4 nodes by degree; per bag: top-64 of the attention row among untaken
// nodes (strict > scan => lowest index wins ties, matching lax.top_k), emitted
// in ascending index order (matching jnp.sort), marking taken globally.
// ---------------------------------------------------------------------------
__global__ void __launch_bounds__(32) select_idx_kernel(
    const float* __restrict__ attention,
    const float* __restrict__ degree,
    int* __restrict__ idx_out)
{
    const int lane = threadIdx.x;          // 0..31, each lane owns 8 nodes
    float dv[8];
#pragma unroll
    for (int j = 0; j < 8; ++j) dv[j] = degree[lane * 8 + j];

    int b4[4];
    for (int i = 0; i < 4; ++i) {
        float best = -INFINITY; int bidx = 0x7fffffff;
#pragma unroll
        for (int j = 0; j < 8; ++j)
            if (dv[j] > best) { best = dv[j]; bidx = lane * 8 + j; }
#pragma unroll
        for (int off = 16; off >= 1; off >>= 1) {
            float ov = __shfl_xor(best, off);
            int   oi = __shfl_xor(bidx, off);
            if (ov > best || (ov == best && oi < bidx)) { best = ov; bidx = oi; }
        }
        b4[i] = bidx;                       // broadcast-identical in all lanes
        if ((bidx >> 3) == lane) dv[bidx & 7] = -INFINITY;
    }

    unsigned tk = 0;                        // per-lane taken mask (8 nodes)
    for (int bag = 0; bag < 4; ++bag) {
        const float* arow = attention + (size_t)b4[bag] * N_NODES;
        float pv[8];
#pragma unroll
        for (int j = 0; j < 8; ++j)
            pv[j] = ((tk >> j) & 1u) ? -INFINITY : arow[lane * 8 + j];

        unsigned sel = 0;
        for (int pick = 0; pick < 64; ++pick) {
            float best = -INFINITY; int bidx = 0x7fffffff;
#pragma unroll
            for (int j = 0; j < 8; ++j)
                if (pv[j] > best) { best = pv[j]; bidx = lane * 8 + j; }
#pragma unroll
            for (int off = 16; off >= 1; off >>= 1) {
                float ov = __shfl_xor(best, off);
                int   oi = __shfl_xor(bidx, off);
                if (ov > best || (ov == best && oi < bidx)) { best = ov; bidx = oi; }
            }
            if ((bidx >> 3) == lane) { pv[bidx & 7] = -INFINITY; sel |= 1u << (bidx & 7); }
        }
        // emit ascending-by-index: exclusive prefix of popcounts over lanes
        int cnt = __popc(sel);
        int inc = cnt;
#pragma unroll
        for (int off = 1; off < 32; off <<= 1) {
            int t = __shfl_up(inc, off);
            if (lane >= off) inc += t;
        }
        int pos = inc - cnt;
        for (int j = 0; j < 8; ++j)
            if ((sel >> j) & 1u) { idx_out[bag * 64 + pos] = lane * 8 + j; ++pos; }
        tk |= sel;
    }
}

// ---------------------------------------------------------------------------
// Kernel 2: fused conv1 + gather + 3x WMMA GEMM + mask + conv2.
// Block = 256 threads = 8 waves; each wave owns 4 batch elements
// (=> 16x64 A-matrix of mx rows, row m = 4*e + g).
// ---------------------------------------------------------------------------
__global__ void __launch_bounds__(256) fused_main_kernel(
    const float* __restrict__ x,
    const int*   __restrict__ idx,
    const float* __restrict__ c1w, const float* __restrict__ c1b,
    const float* __restrict__ W1,  const float* __restrict__ b1,
    const float* __restrict__ W2,  const float* __restrict__ b2,
    const float* __restrict__ Wa,  const float* __restrict__ ba,
    const float* __restrict__ c2w, const float* __restrict__ c2b,
    float* __restrict__ out, int B)
{
    __shared__ __align__(16) float W1s[64 * WSTR];
    __shared__ __align__(16) float W2s[64 * WSTR];
    __shared__ __align__(16) float Was[64 * WSTR];
    __shared__ float b1s[64], b2s[64], bas[64];
    __shared__ float c2ws[48], c2bs[4], c1ws[4];
    __shared__ int   idxs[256];
    __shared__ __align__(16) float wavebuf[8][3 * 16 * WSTR]; // per-wave: rows/masked | mx | h

    const int tid = threadIdx.x;

    // cooperative weight staging (pad row stride 64 -> 66)
    for (int i = tid; i < 64 * 64; i += 256) {
        int n = i >> 6, k = i & 63;
        W1s[n * WSTR + k] = W1[i];
        W2s[n * WSTR + k] = W2[i];
        Was[n * WSTR + k] = Wa[i];
    }
    if (tid < 64) { b1s[tid] = b1[tid]; b2s[tid] = b2[tid]; bas[tid] = ba[tid]; }
    if (tid < 48) c2ws[tid] = c2w[tid];
    if (tid < 4)  c2bs[tid] = c2b[tid];
    if (tid < 3)  c1ws[tid] = c1w[tid];
    if (tid == 3) c1ws[3] = c1b[0];
    idxs[tid] = idx[tid];
    __syncthreads();

    const int wave = tid >> 5, lane = tid & 31;
    float* rowsb = &wavebuf[wave][0];              // 4x256 x-rows, later 16x66 masked
    float* mxs   = &wavebuf[wave][16 * WSTR];      // mx  [16][66]
    float* hs    = &wavebuf[wave][32 * WSTR];      // h   [16][66]
    const int b0 = blockIdx.x * 32 + wave * 4;

    // ---- 1) stage 4 x-rows (coalesced, streaming/NT) ----
#pragma unroll
    for (int e = 0; e < 4; ++e)
#pragma unroll
        for (int i = 0; i < 8; ++i) {
            int t = i * 32 + lane;
            rowsb[e * RSTR + t] =
                __builtin_nontemporal_load(x + (size_t)(b0 + e) * 256 + t);
        }
    wave_wait_lds();

    // ---- 2) conv1 (1->1, k=3, same) + tanh(tanh) -> x_site ----
    {
        const float w0 = c1ws[0], w1 = c1ws[1], w2 = c1ws[2], bb = c1ws[3];
        const size_t site_base = (size_t)B * 256;
#pragma unroll
        for (int e = 0; e < 4; ++e) {
            const float* r = rowsb + e * RSTR;
#pragma unroll
            for (int i = 0; i < 8; ++i) {
                int t = i * 32 + lane;
                float xm = (t > 0)   ? r[t - 1] : 0.f;
                float xc = r[t];
                float xp = (t < 255) ? r[t + 1] : 0.f;
                float y = fmaf(w0, xm, fmaf(w1, xc, fmaf(w2, xp, bb)));
                __builtin_nontemporal_store(fast_tanh(fast_tanh(y)),
                    out + site_base + (size_t)(b0 + e) * 256 + t);
            }
        }
    }

    // ---- 3) gather mx[16][64], row m = e*4 + g ----
#pragma unroll 4
    for (int i = 0; i < 32; ++i) {
        int lin = i * 32 + lane;          // 0..1023
        int m = lin >> 6, j = lin & 63;
        int e = m >> 2,  g = m & 3;
        mxs[m * WSTR + j] = rowsb[e * RSTR + idxs[g * 64 + j]];
    }
    wave_wait_lds();

    // Fragment indexing for V_WMMA_F32_16X16X4_F32 (wave32):
    //   A 16x4 : lane l -> M = l&15, K = 4*ks + 2*(l>>4) + v      (v = vgpr 0/1)
    //   B 4x16 : lane l -> N = l&15, K = 4*ks + 2*(l>>4) + v ; B[k][n] = W[n][k]
    //   C/D    : lane l, vgpr j -> M = j + 8*(l>>4), N = l&15
    const int lm = lane & 15;
    const int hi = lane >> 4;

    // hoist all 16 A-fragments of mx into registers (reused by 4 N-tiles x 2 GEMMs)
    v2f amx[16];
#pragma unroll
    for (int ks = 0; ks < 16; ++ks)
        amx[ks] = *(const v2f*)(mxs + lm * WSTR + ks * 4 + 2 * hi);

    // ---- 4) l1/l2 GEMMs; h = relu(l1) + relu(l2) -> hs ----
#pragma unroll
    for (int nt = 0; nt < 4; ++nt) {
        const int n = nt * 16 + lm;
        v8f c1v, c2v;
        const float bb1 = b1s[n], bb2 = b2s[n];
#pragma unroll
        for (int j = 0; j < 8; ++j) { c1v[j] = bb1; c2v[j] = bb2; }
#pragma unroll
        for (int ks = 0; ks < 16; ++ks) {
            const int kb = ks * 4 + 2 * hi;
            v2f w1f = *(const v2f*)(W1s + n * WSTR + kb);
            v2f w2f = *(const v2f*)(W2s + n * WSTR + kb);
            c1v = __builtin_amdgcn_wmma_f32_16x16x4_f32(false, amx[ks], false, w1f, (short)0, c1v, false, false);
            c2v = __builtin_amdgcn_wmma_f32_16x16x4_f32(false, amx[ks], false, w2f, (short)0, c2v, false, false);
        }
#pragma unroll
        for (int j = 0; j < 8; ++j) {
            int m = j + 8 * hi;
            hs[m * WSTR + n] = fmaxf(c1v[j], 0.f) + fmaxf(c2v[j], 0.f);
        }
    }
    wave_wait_lds();

    // hoist all 16 A-fragments of h
    v2f ah[16];
#pragma unroll
    for (int ks = 0; ks < 16; ++ks)
        ah[ks] = *(const v2f*)(hs + lm * WSTR + ks * 4 + 2 * hi);

    // ---- 5) Aimi GEMM; masked = (sigmoid(.)+1)*mx -> rowsb (x-rows dead now) ----
#pragma unroll
    for (int nt = 0; nt < 4; ++nt) {
        const int n = nt * 16 + lm;
        v8f c3;
        const float bbA = bas[n];
#pragma unroll
        for (int j = 0; j < 8; ++j) c3[j] = bbA;
#pragma unroll
        for (int ks = 0; ks < 16; ++ks) {
            const int kb = ks * 4 + 2 * hi;
            v2f wf = *(const v2f*)(Was + n * WSTR + kb);
            c3 = __builtin_amdgcn_wmma_f32_16x16x4_f32(false, ah[ks], false, wf, (short)0, c3, false, false);
        }
#pragma unroll
        for (int j = 0; j < 8; ++j) {
            int m = j + 8 * hi;
            float sig = fast_sigmoid(c3[j]);
            rowsb[m * WSTR + n] = (sig + 1.f) * mxs[m * WSTR + n];
        }
    }
    wave_wait_lds();

    // ---- 6) conv2 (4->4, k=3, same) + tanh(tanh) -> x_bag (NT stores) ----
#pragma unroll
    for (int e = 0; e < 4; ++e)
#pragma unroll
        for (int c = 0; c < 4; ++c)
#pragma unroll
            for (int h2 = 0; h2 < 2; ++h2) {
                int t = h2 * 32 + lane;
                float acc = c2bs[c];
#pragma unroll
                for (int i = 0; i < 4; ++i) {
                    const float* mr = rowsb + (e * 4 + i) * WSTR;
                    float xm = (t > 0)  ? mr[t - 1] : 0.f;
                    float xc = mr[t];
                    float xp = (t < 63) ? mr[t + 1] : 0.f;
                    acc = fmaf(c2ws[c * 12 + i * 3 + 0], xm, acc);
                    acc = fmaf(c2ws[c * 12 + i * 3 + 1], xc, acc);
                    acc = fmaf(c2ws[c * 12 + i * 3 + 2], xp, acc);
                }
                __builtin_nontemporal_store(fast_tanh(fast_tanh(acc)),
                    out + (size_t)(b0 + e) * 256 + c * 64 + t);
            }
}

// ---------------------------------------------------------------------------
extern "C" void kernel_launch(void* const* d_in, const int* in_sizes, int n_in,
                              void* d_out, int out_size, void* d_ws, size_t ws_size,
                              hipStream_t stream)
{
    const float* attention = (const float*)d_in[0];
    const float* degree    = (const float*)d_in[1];
    const float* x         = (const float*)d_in[2];
    const float* conv1_w   = (const float*)d_in[3];
    const float* conv1_b   = (const float*)d_in[4];
    const float* W1        = (const float*)d_in[5];
    const float* b1        = (const float*)d_in[6];
    const float* W2        = (const float*)d_in[7];
    const float* b2        = (const float*)d_in[8];
    const float* Wa        = (const float*)d_in[9];
    const float* ba        = (const float*)d_in[10];
    const float* conv2_w   = (const float*)d_in[11];
    const float* conv2_b   = (const float*)d_in[12];
    float* out = (float*)d_out;

    const int B = in_sizes[2] / N_NODES;   // x is [B, 256, 1]
    int* idx_ws = (int*)d_ws;              // [4,64] selected indices

    select_idx_kernel<<<1, 32, 0, stream>>>(attention, degree, idx_ws);

    const int blocks = B / 32;             // 8 waves/block, 4 batch elems/wave
    fused_main_kernel<<<blocks, 256, 0, stream>>>(
        x, idx_ws, conv1_w, conv1_b, W1, b1, W2, b2, Wa, ba, conv2_w, conv2_b,
        out, B);
}